// PromptModifier_2293512536321
// MI455X (gfx1250) — compile-verified
//
#include <hip/hip_runtime.h>
#include <hip/hip_bf16.h>

typedef __attribute__((ext_vector_type(2))) float v2f;
typedef __attribute__((ext_vector_type(4))) float v4f;
typedef __attribute__((ext_vector_type(8))) float v8f;

#define B_   256
#define N_   196
#define D_   768
#define P_   1024
#define K_   5
#define L_   8
#define R_   (K_ * L_ + N_)   // 236 rows in output tensor

// ---------------------------------------------------------------------------
// 1) 1/||row||_2 via wave32 reduction (one wave per row)
// ---------------------------------------------------------------------------
__global__ void rnorm_rows(const float* __restrict__ x, float* __restrict__ r) {
    const int row  = blockIdx.x;
    const int lane = threadIdx.x;           // 0..31 (wave32)
    const float* p = x + (long)row * D_;
    float s = 0.f;
    #pragma unroll
    for (int i = lane; i < D_; i += 32) { float v = p[i]; s += v * v; }
    #pragma unroll
    for (int off = 16; off; off >>= 1) s += __shfl_xor(s, off, 32);
    if (lane == 0) r[row] = rsqrtf(fmaxf(s, 1e-12f));
}

// ---------------------------------------------------------------------------
// 2) S[b,p] = (q_b . k_p) * rk[p]   via V_WMMA_F32_16X16X4_F32
//    One wave computes one 16x16 tile; K-loop of 768/4 WMMA ops.
//    A 16x4 layout : lane&15 = M row, (lane>>4) selects K pair {0,1}/{2,3}
//    B 4x16 layout : lane&15 = N col, (lane>>4) selects K pair (mirrors A)
//    C/D layout    : VGPR r holds M = r + 8*(lane>>4), N = lane&15
// ---------------------------------------------------------------------------
__global__ void gemm_qk_wmma(const float* __restrict__ q,   // [B_,D_]
                             const float* __restrict__ k,   // [P_,D_]
                             const float* __restrict__ rk,  // [P_]
                             float* __restrict__ S) {       // [B_,P_]
    const int lane = threadIdx.x;         // 0..31
    const int half = lane >> 4;           // 0 or 1
    const int l15  = lane & 15;
    const int mt   = blockIdx.x;          // 0..B_/16-1
    const int nt   = blockIdx.y;          // 0..P_/16-1

    const int arow = mt * 16 + l15;
    const int bcol = nt * 16 + l15;
    const float* aptr = q + (long)arow * D_ + 2 * half;  // 8B aligned
    const float* bptr = k + (long)bcol * D_ + 2 * half;  // 8B aligned

    v8f c = {};
    #pragma unroll 8
    for (int kk = 0; kk < D_; kk += 4) {
        v2f a = *(const v2f*)(aptr + kk);
        v2f b = *(const v2f*)(bptr + kk);
        // (neg_a, A, neg_b, B, c_mod, C, reuse_a, reuse_b)
        c = __builtin_amdgcn_wmma_f32_16x16x4_f32(
                false, a, false, b, (short)0, c, false, false);
    }

    const int n  = nt * 16 + l15;
    const float sc = rk[n];
    const int m0 = mt * 16 + 8 * half;
    #pragma unroll
    for (int r = 0; r < 8; ++r)
        S[(long)(m0 + r) * P_ + n] = c[r] * sc;
}

// ---------------------------------------------------------------------------
// 3) Per-row top-5 (thread per batch row, strict > keeps lowest index on tie,
//    matching jax.lax.top_k) + block reduction of key_query_sim.
// ---------------------------------------------------------------------------
__global__ void topk_and_sim(const float* __restrict__ S,   // [B_,P_]
                             const float* __restrict__ rq,  // [B_]
                             int* __restrict__ idx,         // [B_,K_]
                             float* __restrict__ simOut) {  // scalar
    __shared__ float red[B_];
    const int b = threadIdx.x;            // blockDim.x == B_ == 256

    float v[K_]; int ix[K_];
    #pragma unroll
    for (int j = 0; j < K_; ++j) { v[j] = -3.402823e38f; ix[j] = 0; }

    const float* row = S + (long)b * P_;
    for (int p = 0; p < P_; ++p) {
        float x = row[p];
        if (x > v[K_ - 1]) {
            int j = K_ - 1;
            while (j > 0 && x > v[j - 1]) { v[j] = v[j - 1]; ix[j] = ix[j - 1]; --j; }
            v[j] = x; ix[j] = p;
        }
    }

    float local = 0.f;
    #pragma unroll
    for (int j = 0; j < K_; ++j) { local += v[j]; idx[b * K_ + j] = ix[j]; }
    local *= rq[b];   // apply 1/||q_b|| (row-constant, ordering-invariant)

    red[b] = local;
    __syncthreads();
    #pragma unroll
    for (int s = B_ / 2; s > 0; s >>= 1) {
        if (b < s) red[b] += red[b + s];
        __syncthreads();
    }
    if (b == 0) *simOut = red[0] / (float)B_;
}

// ---------------------------------------------------------------------------
// 4) Fused gather + concat: out[b, 0:40, :]  = prompts[idx[b, r/8], r%8, :]
//                           out[b, 40:236,:] = output[b, r-40, :]
//    128-bit lanes, nontemporal (streaming) — this 340 MB of traffic is the
//    roofline term (~15 us at 23.3 TB/s); NT keeps it out of L2.
// ---------------------------------------------------------------------------
__global__ void gather_concat(const float* __restrict__ prompts, // [P_,L_,D_]
                              const float* __restrict__ outputT, // [B_,N_,D_]
                              const int* __restrict__ idx,       // [B_,K_]
                              float* __restrict__ out) {         // [B_,R_,D_]
    const int C4 = D_ / 4;  // 192 x 16B per row
    long i = (long)blockIdx.x * blockDim.x + threadIdx.x;
    const long total = (long)B_ * R_ * C4;
    if (i >= total) return;

    const int c   = (int)(i % C4);
    const long t  = i / C4;
    const int row = (int)(t % R_);
    const int b   = (int)(t / R_);

    const v4f* src;
    if (row < K_ * L_) {
        const int p = idx[b * K_ + (row >> 3)];
        src = (const v4f*)(prompts + ((long)p * L_ + (row & 7)) * D_);
    } else {
        src = (const v4f*)(outputT + ((long)b * N_ + (row - K_ * L_)) * D_);
    }
    v4f val = __builtin_nontemporal_load(src + c);
    v4f* dst = (v4f*)(out + ((long)b * R_ + row) * D_);
    __builtin_nontemporal_store(val, dst + c);
}

// ---------------------------------------------------------------------------
extern "C" void kernel_launch(void* const* d_in, const int* in_sizes, int n_in,
                              void* d_out, int out_size, void* d_ws, size_t ws_size,
                              hipStream_t stream) {
    const float* q       = (const float*)d_in[0]; // [256,768]
    const float* outputT = (const float*)d_in[1]; // [256,196,768]
    const float* prompts = (const float*)d_in[2]; // [1024,8,768]
    const float* keys    = (const float*)d_in[3]; // [1024,768]
    float* out = (float*)d_out;

    // workspace layout (floats): S[B*P], rk[P], rq[B], idx[B*K] (as int)
    float* S  = (float*)d_ws;
    float* rk = S + (long)B_ * P_;
    float* rq = rk + P_;
    int*  idx = (int*)(rq + B_);

    rnorm_rows<<<P_, 32, 0, stream>>>(keys, rk);
    rnorm_rows<<<B_, 32, 0, stream>>>(q, rq);

    gemm_qk_wmma<<<dim3(B_ / 16, P_ / 16), 32, 0, stream>>>(q, keys, rk, S);

    topk_and_sim<<<1, B_, 0, stream>>>(S, rq, idx, out + (out_size - 1));

    const long total4 = (long)B_ * R_ * (D_ / 4);
    const int blocks = (int)((total4 + 255) / 256);
    gather_concat<<<blocks, 256, 0, stream>>>(prompts, outputT, idx, out);
}